// DropBlock_66176856096811
// MI455X (gfx1250) — compile-verified
//
#include <hip/hip_runtime.h>

#ifndef __has_builtin
#define __has_builtin(x) 0
#endif

#define BS 7
#define HH 56
#define WW 56
#define HM 50
#define WM 50
#define NPLANES (64 * 256)            // 16384
#define PLANE_ELEMS (HM * WM)         // 2500
#define OUT_PLANE (HH * WW)           // 3136
#define TOTAL_OUT (NPLANES * OUT_PLANE)  // 51,380,224

typedef float f32x4 __attribute__((ext_vector_type(4)));
typedef int   i32x4 __attribute__((ext_vector_type(4)));
typedef i32x4 __attribute__((address_space(1)))* gptr_i32x4;
typedef i32x4 __attribute__((address_space(3)))* lptr_i32x4;

#if __has_builtin(__builtin_amdgcn_global_load_async_to_lds_b128)
#define HAVE_ASYNC_LDS 1
#else
#define HAVE_ASYNC_LDS 0
#endif

__global__ void DropBlock_zero_kernel(unsigned int* c) { *c = 0u; }

// One workgroup per (b,c) plane: stage 50x50 mask_u into LDS (async B128),
// build per-row bitmasks, do separable 7-wide binary dilation with shifts,
// emit 56 uint64 row masks of block_mask + exact popcount.
__global__ __launch_bounds__(256) void DropBlock_mask_kernel(
    const float* __restrict__ mask_u,
    unsigned long long* __restrict__ rowmask,
    unsigned int* __restrict__ count) {
  const float gamma =
      (float)(0.1 / (double)(BS * BS) * (double)(HH * HH) /
              (double)((HH - BS + 1) * (HH - BS + 1)));  // 0.00256

  __shared__ __align__(16) float su[PLANE_ELEMS];
  __shared__ unsigned long long rowdil[HM];
  __shared__ unsigned int bcount;

  const int tid = threadIdx.x;
  const int plane = blockIdx.x;
  const float* src = mask_u + (size_t)plane * PLANE_ELEMS;

  if (tid == 0) bcount = 0u;

  // Stage the 10,000-byte plane into LDS: 625 x 16B transfers.
  for (int v = tid; v < PLANE_ELEMS / 4; v += 256) {
#if HAVE_ASYNC_LDS
    __builtin_amdgcn_global_load_async_to_lds_b128(
        (gptr_i32x4)(src + 4 * v),
        (lptr_i32x4)(su + 4 * v), 0, 0);
#else
    ((f32x4*)su)[v] = ((const f32x4*)src)[v];
#endif
  }
#if HAVE_ASYNC_LDS
#if __has_builtin(__builtin_amdgcn_s_wait_asynccnt)
  __builtin_amdgcn_s_wait_asynccnt(0);
#else
  asm volatile("s_wait_asynccnt 0" ::: "memory");
#endif
#endif
  __syncthreads();

  // Per-row bitmask + horizontal dilation (7-wide OR of shifts).
  if (tid < HM) {
    unsigned long long bits = 0ull;
    for (int j = 0; j < WM; ++j)
      bits |= (su[tid * WM + j] < gamma) ? (1ull << j) : 0ull;
    unsigned long long d = 0ull;
#pragma unroll
    for (int s = 0; s < BS; ++s) d |= bits << s;
    rowdil[tid] = d;  // bits 0..55 valid
  }
  __syncthreads();

  // Vertical dilation + invert + pack + count.
  if (tid < HH) {
    int lo = tid - (BS - 1); if (lo < 0) lo = 0;
    int hi = tid;            if (hi > HM - 1) hi = HM - 1;
    unsigned long long blocked = 0ull;
    for (int i = lo; i <= hi; ++i) blocked |= rowdil[i];
    unsigned long long bm = (~blocked) & ((1ull << WW) - 1ull);
    rowmask[(size_t)plane * HH + tid] = bm;
    atomicAdd(&bcount, (unsigned int)__popcll(bm));
  }
  __syncthreads();
  if (tid == 0) atomicAdd(count, bcount);
}

// Streaming apply: out = x * block_mask * (N / count), B128 nontemporal.
__global__ __launch_bounds__(256) void DropBlock_apply_kernel(
    const float* __restrict__ x,
    const unsigned long long* __restrict__ rowmask,
    const unsigned int* __restrict__ count,
    float* __restrict__ out) {
  __shared__ float s_scale;
  if (threadIdx.x == 0)
    s_scale = (float)TOTAL_OUT / (float)(*count);
  __syncthreads();

  const int idx = blockIdx.x * 256 + threadIdx.x;  // float4 index
  const int P4 = OUT_PLANE / 4;                    // 784
  const int R4 = WW / 4;                           // 14
  const int plane = idx / P4;
  const int rem = idx - plane * P4;
  const int row = rem / R4;
  const int c4 = rem - row * R4;

  unsigned long long m = rowmask[plane * HH + row] >> (c4 * 4);
  f32x4 v = __builtin_nontemporal_load((const f32x4*)x + idx);
  const float s = s_scale;
  f32x4 r;
  r.x = (m & 1ull) ? v.x * s : 0.0f;
  r.y = (m & 2ull) ? v.y * s : 0.0f;
  r.z = (m & 4ull) ? v.z * s : 0.0f;
  r.w = (m & 8ull) ? v.w * s : 0.0f;
  __builtin_nontemporal_store(r, (f32x4*)out + idx);
}

extern "C" void kernel_launch(void* const* d_in, const int* in_sizes, int n_in,
                              void* d_out, int out_size, void* d_ws, size_t ws_size,
                              hipStream_t stream) {
  const float* x = (const float*)d_in[0];        // (64,256,56,56) f32
  const float* mask_u = (const float*)d_in[1];   // (64,256,50,50) f32
  float* out = (float*)d_out;

  // Workspace layout: [0..15] atomic count (uint), [16..] 16384*56 uint64 row masks (7.34 MB).
  unsigned int* count = (unsigned int*)d_ws;
  unsigned long long* rowmask = (unsigned long long*)((char*)d_ws + 16);

  DropBlock_zero_kernel<<<1, 1, 0, stream>>>(count);
  DropBlock_mask_kernel<<<NPLANES, 256, 0, stream>>>(mask_u, rowmask, count);
  DropBlock_apply_kernel<<<TOTAL_OUT / 4 / 256, 256, 0, stream>>>(x, rowmask, count, out);
}